// ts_corr_32160715112508
// MI455X (gfx1250) — compile-verified
//
#include <hip/hip_runtime.h>

typedef __attribute__((ext_vector_type(2))) float v2f;
typedef __attribute__((ext_vector_type(8))) float v8f;

constexpr int kFeat  = 15;   // n
constexpr int kT     = 100;  // series length
constexpr int kD     = 10;   // window length
constexpr int kW     = 10;   // (100-10)/10+1 windows
constexpr int kPairs = 105;  // 15*14/2

// One wave32 per (batch, window) tile; one 320-thread block (10 waves) per
// batch, so each block reads a contiguous 6000B slab of X and writes a
// contiguous 4200B slab of corr.
//
// The normalization is folded into the WMMA inputs: with
//   s_i = rsq(ss_i) * 3/sqrt(10),   a'[i][k] = c[i][k] * s_i,
// the Gram of a' is exactly corr (= 9*sum(c_m c_n) / (10*sqrt(ss_m*ss_n))),
// so the WMMA accumulator is stored directly — no cross-lane std broadcast,
// no per-pair scaling. Gram is symmetric, so the A (16x4) and B (4x16) f32
// operands are the same registers. Row 15 (padding) duplicates row 14: its
// Gram rows/cols are never emitted. Only K-padding (k=10,11) must be zero.
__global__ __launch_bounds__(320)
void ts_corr_wmma_kernel(const float* __restrict__ X,
                         float* __restrict__ out) {
  const int b    = blockIdx.x;
  const int w    = (int)(threadIdx.x >> 5);    // window = wave id in block
  const int lane = (int)(threadIdx.x & 31);
  const int r    = lane & 15;                  // matrix row (M) and column (N)
  const int h    = lane >> 4;                  // lane half selects K sub-chunk
  const int rr   = (r < kFeat) ? r : (kFeat - 1);  // clamp: row 15 dups row 14

  // ---- load this row's 10-sample window (8B-aligned v2f loads) ----
  const float* p = X + (size_t)b * (kFeat * kT) + (size_t)rr * kT + w * kD;
  float f[kD];
#pragma unroll
  for (int d = 0; d < kD; d += 2) {
    v2f t = *(const v2f*)(p + d);
    f[d] = t.x; f[d + 1] = t.y;
  }

  // ---- center; fold 1/(std_m*std_n*D) into the operands ----
  float mean = 0.0f;
#pragma unroll
  for (int d = 0; d < kD; ++d) mean += f[d];
  mean *= (1.0f / kD);
  float c[kD];
  float ss = 0.0f;
#pragma unroll
  for (int d = 0; d < kD; ++d) { c[d] = f[d] - mean; ss += c[d] * c[d]; }
  // s = (1/sqrt(ss)) * 3/sqrt(10):  corr = sum_k (c_m*s_m)*(c_n*s_n)
  const float s = __builtin_amdgcn_rsqf(ss) * 0.94868329805051381f;

  // ---- pack pre-scaled WMMA operands (constant indices -> pure VGPRs) ----
  // A 16x4 f32 layout: lanes 0-15 hold K = {0,1} of chunk, lanes 16-31 K = {2,3}.
  v2f a0, a1, a2;
  a0.x = (h ? c[2] : c[0]) * s;  a0.y = (h ? c[3] : c[1]) * s;  // K 0..3
  a1.x = (h ? c[6] : c[4]) * s;  a1.y = (h ? c[7] : c[5]) * s;  // K 4..7
  a2.x = h ? 0.f : c[8] * s;     a2.y = h ? 0.f : c[9] * s;     // K 8..9 + pad

  v8f acc = {};
  acc = __builtin_amdgcn_wmma_f32_16x16x4_f32(false, a2, false, a2, (short)0, acc, false, false);
  acc = __builtin_amdgcn_wmma_f32_16x16x4_f32(false, a1, false, a1, (short)0, acc, false, false);
  acc = __builtin_amdgcn_wmma_f32_16x16x4_f32(false, a0, false, a0, (short)0, acc, false, false);

  // ---- emit upper-triangle corr values (accumulator IS corr) ----
  // D layout: VGPR v, lane L -> corr[m][n], m = v + 8*h, n = lane&15.
  const int rlim = (r < kFeat) ? r : 0;        // store iff m < rlim
  const size_t obase = (size_t)b * (kPairs * kW) + w;
  const int m0 = 8 * h;
  // triu(k=1) rank of pair (m0, r); advanced by (13 - m) per step
  int pidx = m0 * kFeat - (m0 * (m0 + 1)) / 2 + (r - m0 - 1);
#pragma unroll
  for (int v = 0; v < 8; ++v) {
    const int m = m0 + v;
    if (m < rlim) out[obase + (size_t)pidx * kW] = acc[v];
    pidx += (kFeat - 2) - m;                   // 13 - m
  }
}

extern "C" void kernel_launch(void* const* d_in, const int* in_sizes, int n_in,
                              void* d_out, int out_size, void* d_ws, size_t ws_size,
                              hipStream_t stream) {
  const float* X = (const float*)d_in[0];
  float* out = (float*)d_out;
  const int B = in_sizes[0] / (kFeat * kT);    // 16384 batches
  ts_corr_wmma_kernel<<<B, 320, 0, stream>>>(X, out);  // 1 block per batch
}